// GATLayer_47210280517998
// MI455X (gfx1250) — compile-verified
//
#include <hip/hip_runtime.h>
#include <hip/hip_bf16.h>
#include <float.h>

#define N_NODES 50000
#define N_EDGES 800000
#define IN_F    128
#define HEADS   4
#define OUT_C   16
#define HC      64            // HEADS*OUT_C
#define NEGS    0.2f
#define ENTRIES (N_EDGES + N_NODES)   // edges + self loops

typedef __attribute__((ext_vector_type(2))) float v2f;
typedef __attribute__((ext_vector_type(8))) float v8f;

// ---- order-preserving float <-> uint encoding for atomicMax on floats ----
__device__ __forceinline__ unsigned encF(float f) {
    unsigned u = __float_as_uint(f);
    return (u & 0x80000000u) ? ~u : (u | 0x80000000u);
}
__device__ __forceinline__ float decF(unsigned u) {
    unsigned b = (u & 0x80000000u) ? (u & 0x7FFFFFFFu) : ~u;
    return __uint_as_float(b);
}
#define ENC_NEG_FLT_MAX 0x00800000u   // encF(-FLT_MAX)

// ---------------- K0: init workspace ----------------
__global__ void k_init(float* __restrict__ acc, float* __restrict__ deg,
                       float* __restrict__ eam, float* __restrict__ denom,
                       unsigned* __restrict__ amax) {
    int gid = blockIdx.x * blockDim.x + threadIdx.x;
    if (gid < N_NODES * HC) acc[gid] = 0.0f;
    if (gid < N_NODES) { deg[gid] = 0.0f; eam[gid] = 0.0f; }
    if (gid < N_NODES * HEADS) { denom[gid] = 0.0f; amax[gid] = ENC_NEG_FLT_MAX; }
}

// ---------------- K1: degree + edge_attr sum per dst ----------------
__global__ void k_deg(const int* __restrict__ ei, const float* __restrict__ ea,
                      float* __restrict__ deg, float* __restrict__ easum) {
    int e = blockIdx.x * blockDim.x + threadIdx.x;
    if (e >= N_EDGES) return;
    int d = ei[N_EDGES + e];
    atomicAdd(&deg[d], 1.0f);
    atomicAdd(&easum[d], ea[e]);
}

// ---------------- K2: ea mean (in place) + kedge[h] ----------------
__global__ void k_mean(const float* __restrict__ deg, float* __restrict__ eam,
                       const float* __restrict__ W_edge, const float* __restrict__ att_edge,
                       float* __restrict__ kedge) {
    int n = blockIdx.x * blockDim.x + threadIdx.x;
    if (n < HEADS) {
        float s = 0.0f;
        #pragma unroll
        for (int c = 0; c < OUT_C; ++c) s += W_edge[n * OUT_C + c] * att_edge[n * OUT_C + c];
        kedge[n] = s;
    }
    if (n < N_NODES) eam[n] = eam[n] / fmaxf(deg[n], 1.0f);
}

// ---------------- K3: xw = x @ W via V_WMMA_F32_16X16X4_F32 ----------------
// 8 waves/block, each wave computes a 16x64 output tile.
// W held in LDS with row stride 80 (conflict-free half-wave access).
#define W_LDS_STRIDE 80
__global__ void __launch_bounds__(256)
k_gemm(const float* __restrict__ X, const float* __restrict__ W,
       float* __restrict__ XW) {
    __shared__ float Wl[IN_F * W_LDS_STRIDE];
    for (int i = threadIdx.x; i < IN_F * HC; i += blockDim.x) {
        int r = i >> 6, c = i & 63;
        Wl[r * W_LDS_STRIDE + c] = W[i];
    }
    __syncthreads();

    const int wave = threadIdx.x >> 5;
    const int lane = threadIdx.x & 31;
    const int rowTile = (blockIdx.x * 8 + wave) * 16;
    if (rowTile >= N_NODES) return;

    const int mrow = lane & 15;            // A: row,  B/C: column within tile
    const int koff = (lane >> 4) << 1;     // 0 for lanes 0-15, 2 for lanes 16-31
    int srcRow = rowTile + mrow;
    if (srcRow >= N_NODES) srcRow = N_NODES - 1;   // clamp; stores are guarded
    const float* __restrict__ xr = X + (size_t)srcRow * IN_F;

    v8f a0 = {}, a1 = {}, a2 = {}, a3 = {};
    #pragma unroll 4
    for (int k = 0; k < IN_F; k += 4) {
        // A frag 16x4 f32: VGPR0 = K(koff), VGPR1 = K(koff+1)
        v2f a; a.x = xr[k + koff]; a.y = xr[k + koff + 1];
        const float* w0 = &Wl[(k + koff) * W_LDS_STRIDE];
        const float* w1 = &Wl[(k + koff + 1) * W_LDS_STRIDE];
        // B frag 4x16 f32: N = lane&15, lane-halves split K exactly like A
        v2f b0 = { w0[ 0 + mrow], w1[ 0 + mrow] };
        v2f b1 = { w0[16 + mrow], w1[16 + mrow] };
        v2f b2 = { w0[32 + mrow], w1[32 + mrow] };
        v2f b3 = { w0[48 + mrow], w1[48 + mrow] };
        a0 = __builtin_amdgcn_wmma_f32_16x16x4_f32(false, a, false, b0, (short)0, a0, false, false);
        a1 = __builtin_amdgcn_wmma_f32_16x16x4_f32(false, a, false, b1, (short)0, a1, false, false);
        a2 = __builtin_amdgcn_wmma_f32_16x16x4_f32(false, a, false, b2, (short)0, a2, false, false);
        a3 = __builtin_amdgcn_wmma_f32_16x16x4_f32(false, a, false, b3, (short)0, a3, false, false);
    }
    // C/D layout: VGPR i -> M = i (lanes 0-15) or i+8 (lanes 16-31), N = lane&15
    const int rowAdd = (lane >> 4) * 8;
    #pragma unroll
    for (int i = 0; i < 8; ++i) {
        int r = rowTile + rowAdd + i;
        if (r < N_NODES) {
            float* o = XW + (size_t)r * HC + mrow;
            o[0]  = a0[i];
            o[16] = a1[i];
            o[32] = a2[i];
            o[48] = a3[i];
        }
    }
}

// ---------------- K4: per-node attention logits ----------------
__global__ void k_logits(const float* __restrict__ xw,
                         const float* __restrict__ att_src, const float* __restrict__ att_dst,
                         float* __restrict__ asrc, float* __restrict__ adst) {
    int n = blockIdx.x * blockDim.x + threadIdx.x;
    if (n >= N_NODES) return;
    const float* row = xw + (size_t)n * HC;
    #pragma unroll
    for (int h = 0; h < HEADS; ++h) {
        float s = 0.0f, d = 0.0f;
        #pragma unroll
        for (int c = 0; c < OUT_C; ++c) {
            float v = row[h * OUT_C + c];
            s += v * att_src[h * OUT_C + c];
            d += v * att_dst[h * OUT_C + c];
        }
        asrc[n * HEADS + h] = s;
        adst[n * HEADS + h] = d;
    }
}

// ---------------- K5: alpha = leakyrelu(...), segment max via uint atomicMax ----------------
__global__ void k_alpha(const int* __restrict__ ei, const float* __restrict__ ea,
                        const float* __restrict__ eam,
                        const float* __restrict__ asrc, const float* __restrict__ adst,
                        const float* __restrict__ kedge,
                        float* __restrict__ alpha, unsigned* __restrict__ amax) {
    int e = blockIdx.x * blockDim.x + threadIdx.x;
    if (e >= ENTRIES) return;
    int s, d; float av;
    if (e < N_EDGES) { s = ei[e]; d = ei[N_EDGES + e]; av = ea[e]; }
    else             { s = d = e - N_EDGES;            av = eam[s]; }
    #pragma unroll
    for (int h = 0; h < HEADS; ++h) {
        float a = asrc[s * HEADS + h] + adst[d * HEADS + h] + av * kedge[h];
        a = (a > 0.0f) ? a : NEGS * a;
        alpha[(size_t)e * HEADS + h] = a;
        atomicMax(&amax[d * HEADS + h], encF(a));
    }
}

// ---------------- K6: fused exp + denom + weighted scatter ----------------
__global__ void k_scatter(const int* __restrict__ ei, const float* __restrict__ xw,
                          const float* __restrict__ alpha, const unsigned* __restrict__ amax,
                          float* __restrict__ denom, float* __restrict__ acc) {
    long long gid = (long long)blockIdx.x * blockDim.x + threadIdx.x;
    if (gid >= (long long)ENTRIES * HC) return;
    int e  = (int)(gid >> 6);
    int ch = (int)(gid & 63);
    int h  = ch >> 4;
    int s, d;
    if (e < N_EDGES) { s = ei[e]; d = ei[N_EDGES + e]; }
    else             { s = d = e - N_EDGES; }
    float a  = alpha[(size_t)e * HEADS + h];
    float m  = decF(amax[d * HEADS + h]);
    float ex = __expf(a - m);
    if ((ch & 15) == 0) atomicAdd(&denom[d * HEADS + h], ex);
    atomicAdd(&acc[(size_t)d * HC + ch], ex * xw[(size_t)s * HC + ch]);
}

// ---------------- K7: finalize out = relu(acc/denom + bias) ----------------
__global__ void k_final(const float* __restrict__ acc, const float* __restrict__ denom,
                        const float* __restrict__ bias, float* __restrict__ out) {
    int gid = blockIdx.x * blockDim.x + threadIdx.x;
    if (gid >= N_NODES * HC) return;
    int n = gid >> 6, ch = gid & 63, h = ch >> 4;
    float v = acc[gid] / denom[n * HEADS + h] + bias[ch];
    out[gid] = (v > 0.0f) ? v : 0.0f;
}

extern "C" void kernel_launch(void* const* d_in, const int* in_sizes, int n_in,
                              void* d_out, int out_size, void* d_ws, size_t ws_size,
                              hipStream_t stream) {
    (void)in_sizes; (void)n_in; (void)out_size; (void)ws_size;
    const float* x        = (const float*)d_in[0];
    const int*   ei       = (const int*)  d_in[1];   // [2, E] int32: row0=src, row1=dst
    const float* ea       = (const float*)d_in[2];
    const float* W        = (const float*)d_in[3];
    const float* W_edge   = (const float*)d_in[4];
    const float* att_src  = (const float*)d_in[5];
    const float* att_dst  = (const float*)d_in[6];
    const float* att_edge = (const float*)d_in[7];
    const float* bias     = (const float*)d_in[8];
    float* out = (float*)d_out;

    // workspace carving (floats)
    float* ws    = (float*)d_ws;
    float* xw    = ws;                                  // N*64
    float* acc   = xw   + (size_t)N_NODES * HC;         // N*64
    float* deg   = acc  + (size_t)N_NODES * HC;         // N
    float* eam   = deg  + N_NODES;                      // N   (sum -> mean in place)
    float* asrc  = eam  + N_NODES;                      // N*4
    float* adst  = asrc + (size_t)N_NODES * HEADS;      // N*4
    float* denom = adst + (size_t)N_NODES * HEADS;      // N*4
    unsigned* amax = (unsigned*)(denom + (size_t)N_NODES * HEADS);  // N*4
    float* alpha = (float*)(amax + (size_t)N_NODES * HEADS);        // (E+N)*4
    float* kedge = alpha + (size_t)ENTRIES * HEADS;                 // 4

    const int TB = 256;
    // K0: init
    k_init<<<(N_NODES * HC + TB - 1) / TB, TB, 0, stream>>>(acc, deg, eam, denom, amax);
    // K1: degree / edge_attr sum
    k_deg<<<(N_EDGES + TB - 1) / TB, TB, 0, stream>>>(ei, ea, deg, eam);
    // K2: mean + kedge
    k_mean<<<(N_NODES + TB - 1) / TB, TB, 0, stream>>>(deg, eam, W_edge, att_edge, kedge);
    // K3: GEMM (8 waves * 16 rows = 128 rows per block)
    k_gemm<<<(N_NODES + 127) / 128, 256, 0, stream>>>(x, W, xw);
    // K4: node logits
    k_logits<<<(N_NODES + TB - 1) / TB, TB, 0, stream>>>(xw, att_src, att_dst, asrc, adst);
    // K5: alpha + segment max
    k_alpha<<<(ENTRIES + TB - 1) / TB, TB, 0, stream>>>(ei, ea, eam, asrc, adst, kedge, alpha, amax);
    // K6: fused exp/denom/scatter
    long long tot = (long long)ENTRIES * HC;
    k_scatter<<<(unsigned)((tot + TB - 1) / TB), TB, 0, stream>>>(ei, xw, alpha, amax, denom, acc);
    // K7: finalize
    k_final<<<(N_NODES * HC + TB - 1) / TB, TB, 0, stream>>>(acc, denom, bias, out);
}